// CreatorXSimGCL_7988639170782
// MI455X (gfx1250) — compile-verified
//
#include <hip/hip_runtime.h>
#include <hip/hip_bf16.h>

typedef __attribute__((ext_vector_type(2))) float v2f;
typedef __attribute__((ext_vector_type(8))) float v8f;

#define SIMGCL_EPS 0.2f

__device__ __forceinline__ float signf(float x) {
    return (x > 0.f) ? 1.f : ((x < 0.f) ? -1.f : 0.f);
}

// ---------------------------------------------------------------- zero buffer
__global__ __launch_bounds__(256) void zero_kernel(float4* __restrict__ p, long n4) {
    long i = (long)blockIdx.x * blockDim.x + threadIdx.x;
    if (i < n4) p[i] = make_float4(0.f, 0.f, 0.f, 0.f);
}

// ------------------------------------------- ego = [user ; item+author], out=0
__global__ __launch_bounds__(256) void build_ego_kernel(
    const float4* __restrict__ user, const float4* __restrict__ item,
    const float4* __restrict__ author, const int* __restrict__ i2a,
    float4* __restrict__ ego, float4* __restrict__ out, int U, int N) {
    long t = (long)blockIdx.x * blockDim.x + threadIdx.x;
    long total = (long)N * 16;                 // 16 float4 per 64-wide row
    if (t >= total) return;
    int node = (int)(t >> 4);
    int f    = (int)(t & 15);
    float4 v;
    if (node < U) {
        v = user[(long)node * 16 + f];
    } else {
        int j = node - U;
        float4 a = item[(long)j * 16 + f];
        float4 b = author[(long)i2a[j] * 16 + f];
        v = make_float4(a.x + b.x, a.y + b.y, a.z + b.z, a.w + b.w);
    }
    ego[t] = v;
    out[t] = make_float4(0.f, 0.f, 0.f, 0.f); // d_out is the layer accumulator
}

// ------------------------------------------------- COO SpMM: dst += A @ src
// 16 lanes per edge, float4 per lane; gathers + atomics live in 192MB L2.
__global__ __launch_bounds__(256) void spmm_kernel(
    const int* __restrict__ row, const int* __restrict__ col,
    const float* __restrict__ vals, const float4* __restrict__ src,
    float* __restrict__ dst, long E) {
    long t = (long)blockIdx.x * blockDim.x + threadIdx.x;
    long e = t >> 4;
    if (e >= E) return;
    int f = (int)(t & 15);
    int r = row[e];
    int c = col[e];
    float w = vals[e];
    float4 v = src[(long)c * 16 + f];
    float* d = dst + (long)r * 64 + f * 4;
    unsafeAtomicAdd(d + 0, v.x * w);   // global_atomic_add_f32
    unsafeAtomicAdd(d + 1, v.y * w);
    unsafeAtomicAdd(d + 2, v.z * w);
    unsafeAtomicAdd(d + 3, v.w * w);
}

// --------------- epilogue: row-norm of noise via WMMA Gram diagonal, perturb,
// --------------- accumulate mean into out. One wave handles 16 node rows.
__global__ __launch_bounds__(256) void epilogue_kernel(
    float* __restrict__ buf,            // spmm result, perturbed in place
    const float* __restrict__ noise_k,  // [N,64] this layer's noise
    float* __restrict__ out,            // running mean accumulator
    int N, float outscale) {
    __shared__ float sums[8 * 16];      // 8 waves * 16 rows
    int wave = threadIdx.x >> 5;
    int lane = threadIdx.x & 31;
    int base = (blockIdx.x * 8 + wave) * 16;   // wave-uniform

    if (base < N) {
        int r    = base + (lane & 15);
        int half = lane >> 4;
        const float* nrow = noise_k + (long)r * 64;

        // C = sum_k A(16x4) * A^T(4x16); diagonal = per-row sum of squares.
        // A layout (32-bit 16x4): VGPR0 = K0|K2, VGPR1 = K1|K3 across lane
        // halves, M = lane&15. B = A^T has the identical per-lane data.
        v8f c = {0.f, 0.f, 0.f, 0.f, 0.f, 0.f, 0.f, 0.f};
#pragma unroll
        for (int kk = 0; kk < 16; ++kk) {
            int k0 = kk * 4 + half * 2;
            v2f a;
            a.x = nrow[k0];
            a.y = nrow[k0 + 1];
            c = __builtin_amdgcn_wmma_f32_16x16x4_f32(
                    false, a, false, a, (short)0, c, false, false);
        }
        // Diagonal D[m][m]: lanes 0..7 hold rows 0..7 in c[lane];
        // lanes 24..31 hold rows 8..15 in c[lane-24].
        float d = 0.f;
#pragma unroll
        for (int v = 0; v < 8; ++v)
            if (lane == v || lane == v + 24) d = c[v];
        if (lane < 8)       sums[wave * 16 + lane]        = d;
        else if (lane >= 24) sums[wave * 16 + (lane - 16)] = d;
    }
    __syncthreads();
    if (base >= N) return;

    int r    = base + (lane & 15);
    int half = lane >> 4;
    float n2  = sums[wave * 16 + (lane & 15)];
    float inv = SIMGCL_EPS / fmaxf(sqrtf(n2), 1e-12f);

    float4*       b4 = (float4*)(buf + (long)r * 64);
    const float4* n4 = (const float4*)(noise_k + (long)r * 64);
    float4*       o4 = (float4*)(out + (long)r * 64);
#pragma unroll
    for (int q = 0; q < 8; ++q) {
        int idx = half * 8 + q;     // each lane owns 32 of the 64 columns
        float4 s  = b4[idx];
        float4 nz = n4[idx];
        float4 e;
        e.x = s.x + signf(s.x) * nz.x * inv;
        e.y = s.y + signf(s.y) * nz.y * inv;
        e.z = s.z + signf(s.z) * nz.z * inv;
        e.w = s.w + signf(s.w) * nz.w * inv;
        b4[idx] = e;                // becomes next layer's ego
        float4 o = o4[idx];
        o.x += e.x * outscale;
        o.y += e.y * outscale;
        o.z += e.z * outscale;
        o.w += e.w * outscale;
        o4[idx] = o;
    }
}

extern "C" void kernel_launch(void* const* d_in, const int* in_sizes, int n_in,
                              void* d_out, int out_size, void* d_ws, size_t ws_size,
                              hipStream_t stream) {
    const float* user   = (const float*)d_in[0];
    const float* item   = (const float*)d_in[1];
    const float* author = (const float*)d_in[2];
    const int*   i2a    = (const int*)d_in[3];
    const int*   row    = (const int*)d_in[4];
    const int*   col    = (const int*)d_in[5];
    const float* vals   = (const float*)d_in[6];
    const float* noise  = (const float*)d_in[7];

    const int D = 64;
    int U = in_sizes[0] / D;
    int I = in_sizes[1] / D;
    int N = U + I;
    long E = in_sizes[4];
    int L = in_sizes[7] / (N * D);
    if (L <= 0) L = 1;

    float* out  = (float*)d_out;
    float* bufA = (float*)d_ws;                 // current ego
    float* bufB = bufA + (long)N * D;           // spmm target (needs 2*N*64*4 B ws)

    long n4 = (long)N * (D / 4);
    dim3 blk(256);
    unsigned gN4   = (unsigned)((n4 + 255) / 256);
    unsigned gEdge = (unsigned)((E * 16 + 255) / 256);
    unsigned gEpi  = (unsigned)((N + 127) / 128);   // 8 waves * 16 rows / block

    build_ego_kernel<<<gN4, blk, 0, stream>>>(
        (const float4*)user, (const float4*)item, (const float4*)author, i2a,
        (float4*)bufA, (float4*)out, U, N);

    float* cur = bufA;
    float* nxt = bufB;
    for (int k = 0; k < L; ++k) {
        zero_kernel<<<gN4, blk, 0, stream>>>((float4*)nxt, n4);
        spmm_kernel<<<gEdge, blk, 0, stream>>>(row, col, vals,
                                               (const float4*)cur, nxt, E);
        const float* nk = noise + (long)k * N * D;
        epilogue_kernel<<<gEpi, blk, 0, stream>>>(nxt, nk, out, N, 1.0f / (float)L);
        float* t = cur; cur = nxt; nxt = t;     // perturbed result is next ego
    }
}